// VideoCameraAttention_52596169507404
// MI455X (gfx1250) — compile-verified
//
#include <hip/hip_runtime.h>
#include <hip/hip_bf16.h>

typedef float v2f __attribute__((ext_vector_type(2)));
typedef float v8f __attribute__((ext_vector_type(8)));

#define B_     2
#define T_     8
#define N_     256
#define C_     1024
#define H_     16
#define HD_    64
#define S_IMG  2048        // img tokens per batch
#define S_TOK  2056        // img + cam tokens per batch
#define S_PAD  2064        // padded to multiple of 16
#define MT_    129         // 16-row tiles over S_PAD
#define NQKV   3072
#define NEG_   (-1.0e30f)

// ---------------------------------------------------------------------------
// fp32 WMMA: D[16x16] += A[16x4] * B[4x16], all fp32 (exact vs reference)
// ---------------------------------------------------------------------------
__device__ __forceinline__ v8f wmma4(v2f a, v2f b, v8f c) {
  return __builtin_amdgcn_wmma_f32_16x16x4_f32(false, a, false, b, (short)0, c,
                                               false, false);
}

__device__ __forceinline__ float rowmax16(float x) {
#pragma unroll
  for (int m = 1; m < 16; m <<= 1) x = fmaxf(x, __shfl_xor(x, m, 32));
  return x;
}
__device__ __forceinline__ float rowsum16(float x) {
#pragma unroll
  for (int m = 1; m < 16; m <<= 1) x += __shfl_xor(x, m, 32);
  return x;
}

// ---------------------------------------------------------------------------
// Init kernel:
//  * zero padded KV rows of K   (row-major   [bh][kv_pad][d], kv in 2056..2063)
//  * zero padded KV cols of V^T (d-major     [bh][d][kv_pad], kv in 2056..2063)
//  * zero the shared "zero input row" used by padded A-lanes in the QKV GEMM
// Keeps WMMA inputs NaN-free on the very first (pre-poison) call and makes the
// GEMM A-loads unconditional.
// ---------------------------------------------------------------------------
__global__ void init_ws_kernel(float* __restrict__ k_ws,
                               float* __restrict__ v_ws,
                               float* __restrict__ zrow) {
  int i = blockIdx.x * blockDim.x + threadIdx.x;
  const int NPAD = B_ * H_ * 8 * HD_;            // 16384 pad elements each
  if (i < NPAD) {
    int d  = i & (HD_ - 1);
    int r  = (i >> 6) & 7;
    int bh = i >> 9;
    k_ws[(((size_t)bh * S_PAD) + S_TOK + r) * HD_ + d] = 0.0f;          // row
    v_ws[(((size_t)bh * HD_ + d) * S_PAD) + S_TOK + r] = 0.0f;          // col
  }
  if (i < C_) zrow[i] = 0.0f;
}

// ---------------------------------------------------------------------------
// QKV GEMM (fp32 WMMA) + RoPE epilogue + scatter into per-head Q / fused K,V.
// One wave computes a 16x128 strip; grid covers B * MT_ * (NQKV/128) tasks.
// V is scattered d-major (transposed) for b64 B-loads in the attention stage.
// ---------------------------------------------------------------------------
__global__ void qkv_rope_kernel(const float* __restrict__ img,
                                const float* __restrict__ cam,
                                const float* __restrict__ w_qkv,
                                const float* __restrict__ cos_img,
                                const float* __restrict__ sin_img,
                                const float* __restrict__ cos_cam,
                                const float* __restrict__ sin_cam,
                                float* __restrict__ q_ws,
                                float* __restrict__ k_ws,
                                float* __restrict__ v_ws,
                                const float* __restrict__ zrow) {
  const int lane = threadIdx.x & 31;
  const int wave = threadIdx.x >> 5;
  int task = blockIdx.x * 4 + wave;
  const int NSTRIP = NQKV / 128;                  // 24
  const int ns = task % NSTRIP; task /= NSTRIP;
  const int mt = task % MT_;
  const int b  = task / MT_;

  const int mlane = lane & 15;
  const int koff  = (lane >> 4) << 1;             // 0 or 2
  const int tok   = mt * 16 + mlane;              // A-matrix row for this lane
  // Padded lanes read a dedicated zero row -> unconditional, clause-able loads.
  const float* xrow =
      (tok < S_IMG)  ? img + ((size_t)b * S_IMG + tok) * C_
    : (tok < S_TOK)  ? cam + ((size_t)b * T_ + (tok - S_IMG)) * C_
                     : zrow;

  const int colBase = ns * 128;
  v8f acc[8];
#pragma unroll
  for (int t = 0; t < 8; ++t) acc[t] = (v8f){};

  for (int c = 0; c < C_ / 4; ++c) {              // 256 K-chunks of 4
    v2f a = *(const v2f*)(xrow + 4 * c + koff);
    const int k0 = 4 * c + koff;
    const float* wrow0 = w_qkv + (size_t)k0 * NQKV + colBase + mlane;
    const float* wrow1 = wrow0 + NQKV;
#pragma unroll
    for (int t = 0; t < 8; ++t) {
      v2f bb;
      bb.x = wrow0[t * 16];
      bb.y = wrow1[t * 16];
      acc[t] = wmma4(a, bb, acc[t]);
    }
  }

  // Epilogue: RoPE on q/k sections, scatter into workspace layouts.
  const int sec = colBase >> 10;                  // 0=q, 1=k, 2=v
#pragma unroll
  for (int t = 0; t < 8; ++t) {
    const int col = colBase + t * 16 + mlane;
    const int d   = col & (HD_ - 1);
    const int h   = (col >> 6) & (H_ - 1);
#pragma unroll
    for (int v = 0; v < 8; ++v) {
      float val = acc[t][v];
      float prt = __shfl_xor(val, 1, 32);         // interleaved pair partner
      const int m    = v + ((lane >> 4) << 3);
      const int mtok = mt * 16 + m;
      if (mtok >= S_TOK) continue;
      const bool isImg = (mtok < S_IMG);
      float outv = val;
      if (sec < 2) {
        const int pos = isImg ? mtok : (mtok - S_IMG);
        const float* ct = isImg ? cos_img : cos_cam;
        const float* st = isImg ? sin_img : sin_cam;
        const float cs = ct[pos * HD_ + d];
        const float sn = st[pos * HD_ + d];
        outv = (lane & 1) ? fmaf(val, cs,  prt * sn)    // odd d
                          : fmaf(val, cs, -prt * sn);   // even d
      }
      if (sec == 0) {
        q_ws[(((size_t)(b * H_ + h)) * S_PAD + mtok) * HD_ + d] = outv;
      } else {
        const int kv = isImg ? ((mtok >> 8) * (N_ + 1) + 1 + (mtok & (N_ - 1)))
                             : ((mtok - S_IMG) * (N_ + 1));
        if (sec == 1) {       // K: kv-major rows
          k_ws[(((size_t)(b * H_ + h)) * S_PAD + kv) * HD_ + d] = outv;
        } else {              // V: d-major (transposed) for b64 B-loads
          v_ws[(((size_t)(b * H_ + h)) * HD_ + d) * S_PAD + kv] = outv;
        }
      }
    }
  }
}

// ---------------------------------------------------------------------------
// Flash attention: one wave per (b, h, 16-row q tile). Online softmax,
// QK^T and P*V both on fp32 WMMA; P transposed D->A layout via LDS.
// ---------------------------------------------------------------------------
__global__ void attn_kernel(const float* __restrict__ q_ws,
                            const float* __restrict__ k_ws,
                            const float* __restrict__ v_ws,
                            float* __restrict__ x_ws) {
  const int lane = threadIdx.x & 31;
  const int wave = threadIdx.x >> 5;
  int task = blockIdx.x * 4 + wave;               // B*H*MT_ = 4128, exact
  const int qt = task % MT_; task /= MT_;
  const int h  = task % H_;
  const int b  = task / H_;

  const size_t headOff = (size_t)(b * H_ + h) * S_PAD * HD_;
  const float* Q  = q_ws + headOff;
  const float* K  = k_ws + headOff;
  const float* Vt = v_ws + headOff;               // [d][kv_pad]

  const int mlane = lane & 15;
  const int koff  = (lane >> 4) << 1;
  const bool isCam = (qt == MT_ - 1);

  // Q tile in A-layout: 16 chunks of [16x4]
  v2f qa[16];
  {
    const float* qp = Q + (size_t)(qt * 16 + mlane) * HD_ + koff;
#pragma unroll
    for (int c = 0; c < 16; ++c) qa[c] = *(const v2f*)(qp + 4 * c);
  }

  v8f o[4];
#pragma unroll
  for (int t = 0; t < 4; ++t) o[t] = (v8f){};
  float mrow[8], lrow[8];
#pragma unroll
  for (int v = 0; v < 8; ++v) { mrow[v] = NEG_; lrow[v] = 0.0f; }

  __shared__ float pls[4][256];
  float* myp = pls[wave];

  for (int kt = 0; kt < MT_; ++kt) {
    // ---- S = Q K^T (16 WMMAs over d=64) ----
    v8f s = (v8f){};
    {
      const float* kp = K + (size_t)(kt * 16 + mlane) * HD_ + koff;
#pragma unroll
      for (int c = 0; c < 16; ++c) {
        v2f kb = *(const v2f*)(kp + 4 * c);
        s = wmma4(qa[c], kb, s);
      }
    }
    // ---- mask + scale + online softmax ----
    const int kcol   = kt * 16 + mlane;           // this lane's key index
    const int kframe = kcol / (N_ + 1);
    const bool kvok  = (kcol < S_TOK);
#pragma unroll
    for (int v = 0; v < 8; ++v) {
      const int m = v + ((lane >> 4) << 3);
      float sv = s[v] * 0.125f;                   // HD^-0.5
      bool ok = kvok;
      if (isCam) {
        const int tq = (m < T_) ? m : (T_ - 1);
        ok = ok && (kframe <= tq);
      }
      sv = ok ? sv : NEG_;
      const float tmax = rowmax16(sv);
      const float mn   = fmaxf(mrow[v], tmax);
      const float corr = __expf(mrow[v] - mn);
      const float p    = __expf(sv - mn);
      lrow[v] = lrow[v] * corr + rowsum16(p);
      mrow[v] = mn;
#pragma unroll
      for (int t = 0; t < 4; ++t) o[t][v] *= corr;
      s[v] = p;
    }
    // ---- transpose P (D-layout -> A-layout) through LDS ----
    __syncthreads();                              // WAR vs previous reads
#pragma unroll
    for (int v = 0; v < 8; ++v)
      myp[(v + ((lane >> 4) << 3)) * 16 + mlane] = s[v];
    __syncthreads();                              // RAW before reads
    // ---- O += P V (16 WMMAs, V^T gives contiguous b64 B-loads) ----
#pragma unroll
    for (int ck = 0; ck < 4; ++ck) {
      v2f pa = *(const v2f*)(myp + mlane * 16 + ck * 4 + koff);
      const int kr = kt * 16 + ck * 4 + koff;     // kv pair (kr, kr+1)
#pragma unroll
      for (int dt = 0; dt < 4; ++dt) {
        const int dcol = dt * 16 + mlane;
        v2f vb = *(const v2f*)(Vt + (size_t)dcol * S_PAD + kr);
        o[dt] = wmma4(pa, vb, o[dt]);
      }
    }
  }

  // ---- normalize and store into proj input layout [B][S_PAD][C] ----
#pragma unroll
  for (int v = 0; v < 8; ++v) {
    const int m   = v + ((lane >> 4) << 3);
    const int tok = qt * 16 + m;
    const float inv = 1.0f / lrow[v];
#pragma unroll
    for (int dt = 0; dt < 4; ++dt) {
      if (tok < S_TOK) {
        x_ws[((size_t)b * S_PAD + tok) * C_ + h * HD_ + dt * 16 + mlane] =
            o[dt][v] * inv;
      }
    }
  }
}

// ---------------------------------------------------------------------------
// Output projection (fp32 WMMA) + bias, stores into img_out || cam_out.
// ---------------------------------------------------------------------------
__global__ void proj_kernel(const float* __restrict__ x_ws,
                            const float* __restrict__ w_proj,
                            const float* __restrict__ b_proj,
                            float* __restrict__ out) {
  const int lane = threadIdx.x & 31;
  const int wave = threadIdx.x >> 5;
  int task = blockIdx.x * 4 + wave;               // B * MT_ * 8 = 2064, exact
  const int NSTRIP = C_ / 128;                    // 8
  const int ns = task % NSTRIP; task /= NSTRIP;
  const int mt = task % MT_;
  const int b  = task / MT_;

  const int mlane = lane & 15;
  const int koff  = (lane >> 4) << 1;
  const int tok   = mt * 16 + mlane;              // in-bounds: S_PAD rows
  const float* xrow = x_ws + ((size_t)b * S_PAD + tok) * C_;

  const int colBase = ns * 128;
  v8f acc[8];
#pragma unroll
  for (int t = 0; t < 8; ++t) acc[t] = (v8f){};

  for (int c = 0; c < C_ / 4; ++c) {
    v2f a = *(const v2f*)(xrow + 4 * c + koff);
    const int k0 = 4 * c + koff;
    const float* wrow0 = w_proj + (size_t)k0 * C_ + colBase + mlane;
    const float* wrow1 = wrow0 + C_;
#pragma unroll
    for (int t = 0; t < 8; ++t) {
      v2f bb;
      bb.x = wrow0[t * 16];
      bb.y = wrow1[t * 16];
      acc[t] = wmma4(a, bb, acc[t]);
    }
  }

  const size_t camBase = (size_t)B_ * T_ * N_ * C_;   // img_out elements
#pragma unroll
  for (int t = 0; t < 8; ++t) {
    const int col   = colBase + t * 16 + mlane;
    const float bias = b_proj[col];
#pragma unroll
    for (int v = 0; v < 8; ++v) {
      const int m    = v + ((lane >> 4) << 3);
      const int mtok = mt * 16 + m;
      if (mtok >= S_TOK) continue;
      const float outv = acc[t][v] + bias;
      if (mtok < S_IMG) {
        out[((size_t)b * S_IMG + mtok) * C_ + col] = outv;
      } else {
        out[camBase + ((size_t)b * T_ + (mtok - S_IMG)) * C_ + col] = outv;
      }
    }
  }
}

// ---------------------------------------------------------------------------
extern "C" void kernel_launch(void* const* d_in, const int* in_sizes, int n_in,
                              void* d_out, int out_size, void* d_ws,
                              size_t ws_size, hipStream_t stream) {
  const float* img     = (const float*)d_in[0];
  const float* cam     = (const float*)d_in[1];
  const float* w_qkv   = (const float*)d_in[2];
  const float* w_proj  = (const float*)d_in[3];
  const float* b_proj  = (const float*)d_in[4];
  const float* cos_img = (const float*)d_in[5];
  const float* sin_img = (const float*)d_in[6];
  const float* cos_cam = (const float*)d_in[7];
  const float* sin_cam = (const float*)d_in[8];
  // d_in[9] = attn_mask (recomputed analytically in-kernel)

  const size_t HEADSZ = (size_t)B_ * H_ * S_PAD * HD_;  // 4,227,072 floats
  float* q_ws = (float*)d_ws;
  float* k_ws = q_ws + HEADSZ;
  float* v_ws = k_ws + HEADSZ;                          // stored d-major (V^T)
  float* x_ws = v_ws + HEADSZ;                          // B * S_PAD * C floats
  float* zrow = x_ws + (size_t)B_ * S_PAD * C_;         // C_ zeros

  float* out = (float*)d_out;

  // 1) zero pads + zero-row (NaN safety; enables unconditional A-loads)
  init_ws_kernel<<<(B_ * H_ * 8 * HD_ + 255) / 256, 256, 0, stream>>>(
      k_ws, v_ws, zrow);
  // 2) QKV GEMM + RoPE + scatter: B * 129 * 24 wave-tasks, 4 waves/block
  qkv_rope_kernel<<<(B_ * MT_ * (NQKV / 128)) / 4, 128, 0, stream>>>(
      img, cam, w_qkv, cos_img, sin_img, cos_cam, sin_cam, q_ws, k_ws, v_ws,
      zrow);
  // 3) flash attention: B * H * 129 wave-tasks
  attn_kernel<<<(B_ * H_ * MT_) / 4, 128, 0, stream>>>(q_ws, k_ws, v_ws, x_ws);
  // 4) output projection: B * 129 * 8 wave-tasks
  proj_kernel<<<(B_ * MT_ * (C_ / 128)) / 4, 128, 0, stream>>>(x_ws, w_proj,
                                                               b_proj, out);
}